// DeepHGNN_54872502173933
// MI455X (gfx1250) — compile-verified
//
#include <hip/hip_runtime.h>

#define HIDF 128
#define HEADS 4
#define NGRAPH 256
#define EPG 512
#define ETOT (NGRAPH * EPG)      // 131072 edges
#define NEG_SLOPE 0.2f
#define NMAX (NGRAPH * 256)      // 65536 max nodes

typedef float v2f __attribute__((ext_vector_type(2)));
typedef float v8f __attribute__((ext_vector_type(8)));

// ---------------------------------------------------------------------------
// WMMA fp32 GEMM: C[M,NC] = A[M,K] @ B[K, (ldb cols, use NC of them)] (+bias,relu)
// 4 waves per block, each wave computes one 16x16 tile; block covers 64 rows x 16 cols.
// Weight strip (K x 16) staged in LDS, reused by all 4 waves.
// Requires: M % 64 == 0, NC % 16 == 0, K % 4 == 0, K <= 128.
// ---------------------------------------------------------------------------
__global__ void __launch_bounds__(128)
hgnn_wmma_gemm(const float* __restrict__ A, const float* __restrict__ B,
               const float* __restrict__ bias, float* __restrict__ C,
               int M, int K, int ldb, int NC, int relu) {
  __shared__ float Bs[128 * 16];
  const int tid  = threadIdx.x;
  const int wave = tid >> 5;
  const int lane = tid & 31;
  const int half = lane >> 4;    // 0: lanes 0-15, 1: lanes 16-31
  const int l    = lane & 15;
  const int ntile = blockIdx.y;
  const int mtile = blockIdx.x * 4 + wave;

  // stage B strip (K x 16) into LDS
  for (int t = tid; t < K * 16; t += 128) {
    int kk = t >> 4, cc = t & 15;
    Bs[t] = B[kk * ldb + ntile * 16 + cc];
  }
  __syncthreads();

  const int m0 = mtile * 16;
  const float* Arow = A + (size_t)(m0 + l) * K;

  v8f acc = {};
  for (int k0 = 0; k0 < K; k0 += 4) {
    v2f a, b;
    // A (16x4): lanes 0-15 -> K=k0,k0+1 ; lanes 16-31 -> K=k0+2,k0+3 (M = lane&15)
    a.x = Arow[k0 + 2 * half];
    a.y = Arow[k0 + 2 * half + 1];
    // B (4x16): lanes 0-15 -> K=k0,k0+1 ; lanes 16-31 -> K=k0+2,k0+3 (N = lane&15)
    b.x = Bs[(k0 + 2 * half) * 16 + l];
    b.y = Bs[(k0 + 2 * half + 1) * 16 + l];
    acc = __builtin_amdgcn_wmma_f32_16x16x4_f32(false, a, false, b,
                                                (short)0, acc, false, false);
  }

  const int col = ntile * 16 + l;
  const float bv = bias ? bias[col] : 0.0f;
#pragma unroll
  for (int r = 0; r < 8; ++r) {
    int row = m0 + half * 8 + r;   // VGPR r: lanes 0-15 -> M=r, lanes 16-31 -> M=8+r
    float v = acc[r] + bv;
    if (relu) v = v > 0.0f ? v : 0.0f;
    C[(size_t)row * NC + col] = v;
  }
}

// ---------------------------------------------------------------------------
// small utility kernels
// ---------------------------------------------------------------------------
__global__ void hgnn_fill_f32(float* p, float v, int n) {
  int i = blockIdx.x * blockDim.x + threadIdx.x;
  if (i < n) p[i] = v;
}
__global__ void hgnn_fill_i32(int* p, int v, int n) {
  int i = blockIdx.x * blockDim.x + threadIdx.x;
  if (i < n) p[i] = v;
}

__device__ inline float hgnn_lrelu(float z) { return z > 0.0f ? z : NEG_SLOPE * z; }

__device__ inline void hgnn_atomicMaxF(float* addr, float val) {
  int* ia = (int*)addr;
  int cur = __float_as_int(*addr);
  while (val > __int_as_float(cur)) {
    int prev = atomicCAS(ia, cur, __float_as_int(val));
    if (prev == cur) break;
    cur = prev;
  }
}

// ---------------------------------------------------------------------------
// GCN
// ---------------------------------------------------------------------------
__global__ void hgnn_deg_edges(const int* __restrict__ dst, const int* __restrict__ em,
                               float* __restrict__ deg, int E) {
  int e = blockIdx.x * blockDim.x + threadIdx.x;
  if (e < E && em[e]) atomicAdd(&deg[dst[e]], 1.0f);
}
__global__ void hgnn_deg_to_dis(float* deg, int N) {
  int i = blockIdx.x * blockDim.x + threadIdx.x;
  if (i < N) deg[i] = rsqrtf(fmaxf(deg[i], 1e-12f));
}
__global__ void hgnn_gcn_self(const float* __restrict__ xw, const float* __restrict__ dis,
                              float* __restrict__ out, int N) {
  int i = blockIdx.x * blockDim.x + threadIdx.x;
  if (i < N * HIDF) {
    int n = i >> 7;
    float d = dis[n];
    out[i] = xw[i] * d * d;
  }
}
__global__ void hgnn_gcn_scatter(const float* __restrict__ xw, const float* __restrict__ dis,
                                 const int* __restrict__ src, const int* __restrict__ dst,
                                 const int* __restrict__ em, float* __restrict__ out, int E) {
  int i = blockIdx.x * blockDim.x + threadIdx.x;
  if (i < E * HIDF) {
    int e = i >> 7, f = i & 127;
    if (em[e]) {
      int s = src[e], d = dst[e];
      atomicAdd(&out[d * HIDF + f], xw[s * HIDF + f] * dis[s] * dis[d]);
    }
  }
}
__global__ void hgnn_bias_relu(float* __restrict__ x, const float* __restrict__ b, int N) {
  int i = blockIdx.x * blockDim.x + threadIdx.x;
  if (i < N * HIDF) {
    float v = x[i] + b[i & 127];
    x[i] = v > 0.0f ? v : 0.0f;
  }
}

// ---------------------------------------------------------------------------
// TopK pooling
// ---------------------------------------------------------------------------
__global__ void hgnn_normw(const float* __restrict__ w, float* out) {
  __shared__ float s[128];
  int t = threadIdx.x;
  float v = w[t];
  s[t] = v * v;
  __syncthreads();
  for (int o = 64; o > 0; o >>= 1) { if (t < o) s[t] += s[t + o]; __syncthreads(); }
  if (t == 0) out[0] = sqrtf(s[0]);
}
__global__ void hgnn_score(const float* __restrict__ x, const float* __restrict__ w,
                           const float* __restrict__ normp, float* __restrict__ score, int N) {
  int n = blockIdx.x * blockDim.x + threadIdx.x;
  if (n < N) {
    float dot = 0.0f;
    const float* xr = x + (size_t)n * HIDF;
    for (int f = 0; f < HIDF; ++f) dot += xr[f] * w[f];
    score[n] = tanhf(dot / normp[0]);
  }
}
// one block (256 threads) per graph; bitonic sort descending, emit table/vals/perm
__global__ void __launch_bounds__(256)
hgnn_topk(const float* __restrict__ score, int n, int k,
          int* __restrict__ table, float* __restrict__ vals, int* __restrict__ perm) {
  __shared__ float sv[256];
  __shared__ int   si[256];
  int g = blockIdx.x, t = threadIdx.x;
  if (t < n) table[g * n + t] = -1;
  sv[t] = (t < n) ? score[g * n + t] : -__builtin_inff();
  si[t] = t;
  for (int ksz = 2; ksz <= 256; ksz <<= 1) {
    for (int j = ksz >> 1; j > 0; j >>= 1) {
      __syncthreads();
      int ixj = t ^ j;
      if (ixj > t) {
        bool desc = ((t & ksz) == 0);
        float av = sv[t], bv = sv[ixj];
        bool sw = desc ? (av < bv) : (av > bv);
        if (sw) {
          sv[t] = bv; sv[ixj] = av;
          int tmp = si[t]; si[t] = si[ixj]; si[ixj] = tmp;
        }
      }
    }
  }
  __syncthreads();
  if (t < k) {
    int old = si[t];
    table[g * n + old] = g * k + t;
    vals[g * k + t]    = sv[t];
    perm[g * k + t]    = g * n + old;
  }
}
__global__ void hgnn_gather(const float* __restrict__ x_old, const int* __restrict__ perm,
                            const float* __restrict__ vals, float* __restrict__ x_new, int Nn) {
  int i = blockIdx.x * blockDim.x + threadIdx.x;
  if (i < Nn * HIDF) {
    int n = i >> 7, f = i & 127;
    x_new[i] = x_old[perm[n] * HIDF + f] * vals[n];
  }
}
__global__ void hgnn_remap_edges(const int* __restrict__ src, const int* __restrict__ dst,
                                 const int* __restrict__ em, const int* __restrict__ table,
                                 int* __restrict__ src2, int* __restrict__ dst2,
                                 int* __restrict__ em2, int E) {
  int e = blockIdx.x * blockDim.x + threadIdx.x;
  if (e < E) {
    int m = em[e], s2 = 0, d2 = 0;
    if (m) {
      s2 = table[src[e]];
      d2 = table[dst[e]];
      if (s2 < 0 || d2 < 0) { m = 0; s2 = 0; d2 = 0; }
    }
    src2[e] = s2; dst2[e] = d2; em2[e] = m;
  }
}

// ---------------------------------------------------------------------------
// GATv2 (per head): self-loop logit initializes the per-node max (always finite)
// ---------------------------------------------------------------------------
__global__ void hgnn_gat_self(const float* __restrict__ xl, const float* __restrict__ xr,
                              const float* __restrict__ att,
                              float* __restrict__ selfl, float* __restrict__ mx, int N) {
  int wave = threadIdx.x >> 5, lane = threadIdx.x & 31;
  int node = blockIdx.x * 4 + wave;
  if (node >= N) return;
  float s = 0.0f;
  for (int f = lane; f < HIDF; f += 32)
    s += att[f] * hgnn_lrelu(xl[node * HIDF + f] + xr[node * HIDF + f]);
  for (int o = 16; o > 0; o >>= 1) s += __shfl_xor(s, o, 32);
  if (lane == 0) { selfl[node] = s; mx[node] = s; }
}
__global__ void hgnn_gat_edge_logit(const float* __restrict__ xl, const float* __restrict__ xr,
                                    const float* __restrict__ att,
                                    const int* __restrict__ src, const int* __restrict__ dst,
                                    const int* __restrict__ em,
                                    float* __restrict__ logit, float* __restrict__ mx, int E) {
  int wave = threadIdx.x >> 5, lane = threadIdx.x & 31;
  int e = blockIdx.x * 4 + wave;
  if (e >= E || !em[e]) return;
  int s = src[e], d = dst[e];
  float acc = 0.0f;
  for (int f = lane; f < HIDF; f += 32)
    acc += att[f] * hgnn_lrelu(xl[s * HIDF + f] + xr[d * HIDF + f]);
  for (int o = 16; o > 0; o >>= 1) acc += __shfl_xor(acc, o, 32);
  if (lane == 0) {
    logit[e] = acc;
    hgnn_atomicMaxF(&mx[d], acc);
  }
}
__global__ void hgnn_gat_den_init(float* __restrict__ selfl, const float* __restrict__ mx,
                                  float* __restrict__ den, int N) {
  int n = blockIdx.x * blockDim.x + threadIdx.x;
  if (n < N) {
    float a = expf(selfl[n] - mx[n]);
    selfl[n] = a;        // selfl now holds self attention numerator
    den[n] = a;
  }
}
__global__ void hgnn_gat_edge_exp(const int* __restrict__ dst, const int* __restrict__ em,
                                  const float* __restrict__ mx, float* __restrict__ logit,
                                  float* __restrict__ den, int E) {
  int e = blockIdx.x * blockDim.x + threadIdx.x;
  if (e < E && em[e]) {
    float a = expf(logit[e] - mx[dst[e]]);
    logit[e] = a;
    atomicAdd(&den[dst[e]], a);
  }
}
__global__ void hgnn_gat_acc_self(const float* __restrict__ xl, const float* __restrict__ selfa,
                                  const float* __restrict__ den, float* __restrict__ acc, int N) {
  int i = blockIdx.x * blockDim.x + threadIdx.x;
  if (i < N * HIDF) {
    int n = i >> 7;
    acc[i] += (selfa[n] / den[n]) * xl[i] * (1.0f / HEADS);
  }
}
__global__ void hgnn_gat_acc_edge(const float* __restrict__ xl,
                                  const int* __restrict__ src, const int* __restrict__ dst,
                                  const int* __restrict__ em, const float* __restrict__ logit,
                                  const float* __restrict__ den, float* __restrict__ acc, int E) {
  int i = blockIdx.x * blockDim.x + threadIdx.x;
  if (i < E * HIDF) {
    int e = i >> 7, f = i & 127;
    if (em[e]) {
      int s = src[e], d = dst[e];
      atomicAdd(&acc[d * HIDF + f],
                (logit[e] / den[d]) * xl[s * HIDF + f] * (1.0f / HEADS));
    }
  }
}

// ---------------------------------------------------------------------------
// readout + head
// ---------------------------------------------------------------------------
__global__ void hgnn_pool(const float* __restrict__ x, float* __restrict__ g, int n) {
  int i = blockIdx.x * blockDim.x + threadIdx.x;   // (graph, feature)
  if (i < NGRAPH * HIDF) {
    int gr = i >> 7, f = i & 127;
    float s = 0.0f, m = -__builtin_inff();
    for (int j = 0; j < n; ++j) {
      float v = x[((size_t)gr * n + j) * HIDF + f];
      s += v;
      m = fmaxf(m, v);
    }
    g[i] = s * (1.0f + 1.0f / (float)n) + m;       // mean + max + sum
  }
}
__global__ void hgnn_fc2(const float* __restrict__ h, const float* __restrict__ W,
                         const float* __restrict__ b, float* __restrict__ out) {
  int i = blockIdx.x * blockDim.x + threadIdx.x;   // (graph, out)
  if (i < NGRAPH * 3) {
    int g = i / 3, o = i % 3;
    float acc = b[o];
    for (int kk = 0; kk < 64; ++kk) acc += h[g * 64 + kk] * W[kk * 3 + o];
    out[i] = acc;
  }
}

// ---------------------------------------------------------------------------
// host orchestration
// ---------------------------------------------------------------------------
static inline int cdiv(int a, int b) { return (a + b - 1) / b; }

extern "C" void kernel_launch(void* const* d_in, const int* in_sizes, int n_in,
                              void* d_out, int out_size, void* d_ws, size_t ws_size,
                              hipStream_t stream) {
  const float* x0     = (const float*)d_in[0];
  const float* gcn_W  = (const float*)d_in[1];
  const float* gcn_b  = (const float*)d_in[2];
  const float* gat_Wl = (const float*)d_in[3];
  const float* gat_Wr = (const float*)d_in[4];
  const float* gat_at = (const float*)d_in[5];
  const float* gat_b  = (const float*)d_in[6];
  const float* pool_w = (const float*)d_in[7];
  const float* fc1_W  = (const float*)d_in[8];
  const float* fc1_b  = (const float*)d_in[9];
  const float* fc2_W  = (const float*)d_in[10];
  const float* fc2_b  = (const float*)d_in[11];
  const int*   eidx   = (const int*)d_in[12];

  // workspace carve
  char* base = (char*)d_ws;
  size_t off = 0;
  auto carve = [&](size_t bytes) {
    void* p = base + off;
    off += (bytes + 255) & ~(size_t)255;
    return p;
  };
  const size_t BIG = (size_t)NMAX * HIDF * sizeof(float);
  float* P0 = (float*)carve(BIG);   // x_cur
  float* P1 = (float*)carve(BIG);   // gcn xw / gat xl
  float* P2 = (float*)carve(BIG);   // gcn out / gat xr
  float* P3 = (float*)carve(BIG);   // pooled x (GAT input)
  float* P4 = (float*)carve(BIG);   // gat accumulator
  float* deg   = (float*)carve((size_t)NMAX * 4);
  float* score = (float*)carve((size_t)NMAX * 4);
  float* mx    = (float*)carve((size_t)NMAX * 4);
  float* selfl = (float*)carve((size_t)NMAX * 4);
  float* den   = (float*)carve((size_t)NMAX * 4);
  float* vals  = (float*)carve((size_t)NMAX * 4);
  float* normw = (float*)carve(256);
  int* table = (int*)carve((size_t)NMAX * 4);
  int* perm  = (int*)carve((size_t)NMAX * 4);
  int* srcA = (int*)carve((size_t)ETOT * 4);
  int* dstA = (int*)carve((size_t)ETOT * 4);
  int* emA  = (int*)carve((size_t)ETOT * 4);
  int* srcB = (int*)carve((size_t)ETOT * 4);
  int* dstB = (int*)carve((size_t)ETOT * 4);
  int* emB  = (int*)carve((size_t)ETOT * 4);
  float* logit = (float*)carve((size_t)ETOT * 4);
  float* gbuf  = (float*)carve((size_t)NGRAPH * HIDF * 4);
  float* hbuf  = (float*)carve((size_t)NGRAPH * 64 * 4);
  (void)in_sizes; (void)n_in; (void)out_size; (void)ws_size;

  const int E = ETOT;
  hipMemcpyAsync(P0, x0, (size_t)NMAX * HIDF * sizeof(float),
                 hipMemcpyDeviceToDevice, stream);
  hipMemcpyAsync(srcA, eidx,      (size_t)E * 4, hipMemcpyDeviceToDevice, stream);
  hipMemcpyAsync(dstA, eidx + E,  (size_t)E * 4, hipMemcpyDeviceToDevice, stream);
  hgnn_fill_i32<<<cdiv(E, 256), 256, 0, stream>>>(emA, 1, E);

  int* srcC = srcA; int* dstC = dstA; int* emC = emA;
  int* srcN = srcB; int* dstN = dstB; int* emN = emB;

  int n = 256;
  for (int i = 0; i < 6; ++i) {
    const int N = NGRAPH * n;
    // ---- GCN: xw = x @ W ----
    {
      dim3 grid(N / 64, HIDF / 16);
      hgnn_wmma_gemm<<<grid, 128, 0, stream>>>(P0, gcn_W + (size_t)i * HIDF * HIDF,
                                               nullptr, P1, N, HIDF, HIDF, HIDF, 0);
    }
    hgnn_fill_f32<<<cdiv(N, 256), 256, 0, stream>>>(deg, 1.0f, N);   // self loops
    hgnn_deg_edges<<<cdiv(E, 256), 256, 0, stream>>>(dstC, emC, deg, E);
    hgnn_deg_to_dis<<<cdiv(N, 256), 256, 0, stream>>>(deg, N);
    hgnn_gcn_self<<<cdiv(N * HIDF, 256), 256, 0, stream>>>(P1, deg, P2, N);
    hgnn_gcn_scatter<<<cdiv(E * HIDF, 256), 256, 0, stream>>>(P1, deg, srcC, dstC, emC, P2, E);
    hgnn_bias_relu<<<cdiv(N * HIDF, 256), 256, 0, stream>>>(P2, gcn_b + (size_t)i * HIDF, N);

    if (i == 5) break;   // last layer: no pool / GAT

    // ---- TopK pool ----
    const int k = (4 * n + 4) / 5;
    const int Nn = NGRAPH * k;
    hgnn_normw<<<1, 128, 0, stream>>>(pool_w + (size_t)i * HIDF, normw);
    hgnn_score<<<cdiv(N, 256), 256, 0, stream>>>(P2, pool_w + (size_t)i * HIDF, normw, score, N);
    hgnn_topk<<<NGRAPH, 256, 0, stream>>>(score, n, k, table, vals, perm);
    hgnn_gather<<<cdiv(Nn * HIDF, 256), 256, 0, stream>>>(P2, perm, vals, P3, Nn);
    hgnn_remap_edges<<<cdiv(E, 256), 256, 0, stream>>>(srcC, dstC, emC, table,
                                                       srcN, dstN, emN, E);
    { int* t;
      t = srcC; srcC = srcN; srcN = t;
      t = dstC; dstC = dstN; dstN = t;
      t = emC;  emC  = emN;  emN  = t; }

    // ---- GATv2 (per head) ----
    hgnn_fill_f32<<<cdiv(Nn * HIDF, 256), 256, 0, stream>>>(P4, 0.0f, Nn * HIDF);
    for (int h = 0; h < HEADS; ++h) {
      const float* Wl = gat_Wl + (size_t)i * HIDF * (HEADS * HIDF) + h * HIDF;
      const float* Wr = gat_Wr + (size_t)i * HIDF * (HEADS * HIDF) + h * HIDF;
      const float* at = gat_at + ((size_t)i * HEADS + h) * HIDF;
      dim3 grid(Nn / 64, HIDF / 16);
      hgnn_wmma_gemm<<<grid, 128, 0, stream>>>(P3, Wl, nullptr, P1,
                                               Nn, HIDF, HEADS * HIDF, HIDF, 0);
      hgnn_wmma_gemm<<<grid, 128, 0, stream>>>(P3, Wr, nullptr, P2,
                                               Nn, HIDF, HEADS * HIDF, HIDF, 0);
      hgnn_gat_self<<<cdiv(Nn, 4), 128, 0, stream>>>(P1, P2, at, selfl, mx, Nn);
      hgnn_gat_edge_logit<<<cdiv(E, 4), 128, 0, stream>>>(P1, P2, at, srcC, dstC, emC,
                                                          logit, mx, E);
      hgnn_gat_den_init<<<cdiv(Nn, 256), 256, 0, stream>>>(selfl, mx, den, Nn);
      hgnn_gat_edge_exp<<<cdiv(E, 256), 256, 0, stream>>>(dstC, emC, mx, logit, den, E);
      hgnn_gat_acc_self<<<cdiv(Nn * HIDF, 256), 256, 0, stream>>>(P1, selfl, den, P4, Nn);
      hgnn_gat_acc_edge<<<cdiv(E * HIDF, 256), 256, 0, stream>>>(P1, srcC, dstC, emC,
                                                                 logit, den, P4, E);
    }
    hgnn_bias_relu<<<cdiv(Nn * HIDF, 256), 256, 0, stream>>>(P4, gat_b + (size_t)i * HIDF, Nn);

    { float* t = P0; P0 = P4; P4 = t; }   // GAT output becomes next x
    n = k;
  }

  // ---- readout: mean + max + sum pool over n nodes per graph (x in P2) ----
  hgnn_pool<<<cdiv(NGRAPH * HIDF, 256), 256, 0, stream>>>(P2, gbuf, n);

  // ---- fc1 (WMMA, 256x128 @ 128x64, bias+relu) ----
  {
    dim3 grid(NGRAPH / 64, 64 / 16);
    hgnn_wmma_gemm<<<grid, 128, 0, stream>>>(gbuf, fc1_W, fc1_b, hbuf,
                                             NGRAPH, HIDF, 64, 64, 1);
  }
  // ---- fc2 (256x64 @ 64x3) ----
  hgnn_fc2<<<cdiv(NGRAPH * 3, 256), 256, 0, stream>>>(hbuf, fc2_W, fc2_b, (float*)d_out);
}